// ModelTorch_67293547594182
// MI455X (gfx1250) — compile-verified
//
#include <hip/hip_runtime.h>
#include <hip/hip_bf16.h>

typedef __attribute__((ext_vector_type(2))) float v2f;
typedef __attribute__((ext_vector_type(4))) float v4f;
typedef __attribute__((ext_vector_type(8))) float v8f;
typedef int stage_v4i __attribute__((vector_size(16)));   // matches builtin proto

#define B_    16
#define H_    32
#define KVH_  8
#define D_    128
#define PAGE_ 16
#define LMAX_ 1024
#define GROUPS_ (H_ / KVH_)                 // 4 heads per KV head

// strides in floats inside paged_kv_cache: (Bp, 2, KVH, PAGE, D)
#define PAGE_STRIDE_ (2 * KVH_ * PAGE_ * D_)   // 32768
#define KV_STRIDE_   (KVH_ * PAGE_ * D_)       // 16384
#define KVH_STRIDE_  (PAGE_ * D_)              // 2048

// ---- coalesced row staging: 16B per lane, one full 512B row per op ----------
#if __has_builtin(__builtin_amdgcn_global_load_async_to_lds_b128)
#define STAGE16(dst, src)                                                      \
    __builtin_amdgcn_global_load_async_to_lds_b128(                            \
        (__attribute__((address_space(1))) stage_v4i*)(src),                   \
        (__attribute__((address_space(3))) stage_v4i*)(dst), 0, 0)
#if __has_builtin(__builtin_amdgcn_s_wait_asynccnt)
#define STAGE_WAIT()                                                           \
    do { __builtin_amdgcn_s_wait_asynccnt(0);                                  \
         asm volatile("" ::: "memory"); } while (0)
#else
#define STAGE_WAIT()                                                           \
    asm volatile("s_wait_asynccnt 0x0" ::: "memory")
#endif
#else
#define STAGE16(dst, src) (*(v4f*)(dst) = *(const v4f*)(src))
#define STAGE_WAIT() asm volatile("" ::: "memory")
#endif

__global__ __launch_bounds__(256)
void sparse_paged_attn_kernel(const float* __restrict__ q,
                              const float* __restrict__ kv_cache,
                              const int*   __restrict__ indptr,
                              const int*   __restrict__ pindices,
                              const int*   __restrict__ sind,
                              const int*   __restrict__ snnz,
                              float*       __restrict__ out)
{
    __shared__ float    q_sh[D_];
    __shared__ float    sc[LMAX_];          // scores, then softmax weights
    __shared__ unsigned ro[LMAX_];          // per-row float-offset into kv_cache
    __shared__ float    red[8];
    __shared__ float    kbuf[8][16 * D_];   // per-wave staged K tile (8 KB/wave)

    const int h    = blockIdx.x;
    const int b    = blockIdx.y;
    const int tid  = threadIdx.x;
    const int lane = tid & 31;
    const int wav  = tid >> 5;              // 8 waves
    const int mrow = lane & 15;             // M / N position within wave
    const int kh   = lane >> 4;             // K half (0: K=0,1 ; 1: K=2,3)
    const int kvh  = h / GROUPS_;

    const int    nnz   = snnz[b * H_ + h];
    const int    ip0   = indptr[b];
    const size_t qoff  = (size_t)(b * H_ + h) * D_;
    const int    sbase = (b * H_ + h) * LMAX_;
    const float  scale = 0.08838834764831845f;   // 1/sqrt(128)

    if (tid < D_) q_sh[tid] = q[qoff + tid];
    __syncthreads();

    // ---------------- Phase 1: scores = (K_gather . q) * scale via f32 WMMA ----
    // Each wave owns tiles of 16 l-values; K rows staged coalesced into LDS.
    for (int t = wav; t < LMAX_ / 16; t += 8) {
        const int lbase = t * 16;
        if (lbase >= nnz) {                 // fully invalid tile: fillers only
            if (mrow == 0) {
                #pragma unroll
                for (int v = 0; v < 8; ++v) sc[lbase + kh * 8 + v] = -1e30f;
            }
            continue;
        }
        const int l   = lbase + mrow;
        const int si  = sind[sbase + l];                 // token index
        const int pid = pindices[ip0 + (si >> 4)];       // physical page
        const unsigned roff = (unsigned)pid * PAGE_STRIDE_
                            + (unsigned)kvh * KVH_STRIDE_
                            + (unsigned)(si & (PAGE_ - 1)) * D_;
        if (kh == 0) ro[l] = roff;                       // cache for V pass

        // Stage 16 rows: one fully-coalesced 512B transfer per row.
        float* kb = kbuf[wav];
        #pragma unroll
        for (int i = 0; i < 16; ++i) {
            const unsigned rp = __shfl(roff, i, 32);     // row i's base offset
            const float* src = kv_cache + (size_t)rp + lane * 4;   // 16B / lane
            float*       dst = kb + i * D_ + lane * 4;
            STAGE16(dst, src);
        }
        STAGE_WAIT();

        // WMMA: A = staged K rows (lane reads its row from LDS), B = q bcast.
        const float* kp = kb + mrow * D_;
        v8f acc0 = {}; v8f acc1 = {};
        #pragma unroll 8
        for (int c = 0; c < 32; c += 2) {
            const int col0 = 4 * c + 2 * kh;
            const int col1 = col0 + 4;
            v2f a0 = *(const v2f*)(kp   + col0);
            v2f b0 = *(const v2f*)(q_sh + col0);
            acc0 = __builtin_amdgcn_wmma_f32_16x16x4_f32(
                       false, a0, false, b0, (short)0, acc0, false, false);
            v2f a1 = *(const v2f*)(kp   + col1);
            v2f b1 = *(const v2f*)(q_sh + col1);
            acc1 = __builtin_amdgcn_wmma_f32_16x16x4_f32(
                       false, a1, false, b1, (short)0, acc1, false, false);
        }
        // D[m][n] identical across n; lane 0 holds rows 0..7, lane 16 rows 8..15
        if (mrow == 0) {
            #pragma unroll
            for (int v = 0; v < 8; ++v) {
                const int li = lbase + kh * 8 + v;
                const float s = (acc0[v] + acc1[v]) * scale;
                sc[li] = (li < nnz) ? s : -1e30f;
            }
        }
    }
    __syncthreads();

    // ---------------- Phase 2: masked softmax (exact fp32) ---------------------
    float m = -3.0e38f;
    for (int i = tid; i < LMAX_; i += 256) m = fmaxf(m, sc[i]);
    #pragma unroll
    for (int o = 16; o > 0; o >>= 1) m = fmaxf(m, __shfl_xor(m, o, 32));
    if (lane == 0) red[wav] = m;
    __syncthreads();
    float maxv = red[0];
    #pragma unroll
    for (int w = 1; w < 8; ++w) maxv = fmaxf(maxv, red[w]);

    float s = 0.0f;
    for (int i = tid; i < LMAX_; i += 256) {
        const float e = __expf(sc[i] - maxv);    // invalid slots -> 0
        sc[i] = e;
        s += e;
    }
    #pragma unroll
    for (int o = 16; o > 0; o >>= 1) s += __shfl_xor(s, o, 32);
    __syncthreads();                             // protect red[] reads above
    if (lane == 0) red[wav] = s;
    __syncthreads();
    float sum = 0.0f;
    #pragma unroll
    for (int w = 0; w < 8; ++w) sum += red[w];
    const float inv = (nnz > 0) ? (1.0f / sum) : 0.0f;

    // ---------------- Phase 3: out = softmax . V_gather via f32 WMMA -----------
    // Wave w owns head dims [16w, 16w+16); K-dim = 4 l-values per WMMA.
    const float* vbase   = kv_cache + KV_STRIDE_;   // V part (dim1 == 1)
    const int    dimbase = wav * 16;
    v8f oacc0 = {}; v8f oacc1 = {};
    const int nch = (nnz + 3) >> 2;                 // chunks of 4 rows
    for (int ch = 0; ch < nch; ch += 2) {
        {
            const int l0 = ch * 4 + 2 * kh;
            const unsigned ra = ro[l0], rb = ro[l0 + 1];
            v2f bm; bm[0] = vbase[(size_t)ra + dimbase + mrow];
                    bm[1] = vbase[(size_t)rb + dimbase + mrow];
            v2f ap; ap[0] = sc[l0]; ap[1] = sc[l0 + 1];   // weights bcast over M
            oacc0 = __builtin_amdgcn_wmma_f32_16x16x4_f32(
                        false, ap, false, bm, (short)0, oacc0, false, false);
        }
        if (ch + 1 < nch) {                         // uniform condition
            const int l0 = (ch + 1) * 4 + 2 * kh;
            const unsigned ra = ro[l0], rb = ro[l0 + 1];
            v2f bm; bm[0] = vbase[(size_t)ra + dimbase + mrow];
                    bm[1] = vbase[(size_t)rb + dimbase + mrow];
            v2f ap; ap[0] = sc[l0]; ap[1] = sc[l0 + 1];
            oacc1 = __builtin_amdgcn_wmma_f32_16x16x4_f32(
                        false, ap, false, bm, (short)0, oacc1, false, false);
        }
    }
    // All D rows identical; lanes 0..15 (kh==0) store 16 contiguous dims.
    if (kh == 0) {
        const float val = (oacc0[0] + oacc1[0]) * inv;
        out[qoff + dimbase + mrow] = (nnz > 0) ? val : 0.0f;
    }
}

extern "C" void kernel_launch(void* const* d_in, const int* in_sizes, int n_in,
                              void* d_out, int out_size, void* d_ws, size_t ws_size,
                              hipStream_t stream) {
    (void)in_sizes; (void)n_in; (void)out_size; (void)d_ws; (void)ws_size;
    const float* q        = (const float*)d_in[0];
    const float* kv_cache = (const float*)d_in[1];
    const int*   indptr   = (const int*)d_in[2];
    const int*   pindices = (const int*)d_in[3];
    const int*   sind     = (const int*)d_in[4];
    const int*   snnz     = (const int*)d_in[5];
    float*       out      = (float*)d_out;

    dim3 grid(H_, B_);        // one workgroup per (b, h)
    dim3 block(256);          // 8 wave32 waves
    sparse_paged_attn_kernel<<<grid, block, 0, stream>>>(
        q, kv_cache, indptr, pindices, sind, snnz, out);
}